// CorrespondenceLoss_2439541424695
// MI455X (gfx1250) — compile-verified
//
#include <hip/hip_runtime.h>
#include <hip/hip_bf16.h>
#include <math.h>

// ---------------------------------------------------------------------------
// CorrespondenceLoss on MI455X (gfx1250)
//   - 3x Chamfer + 3x Sinkhorn-EMD (eps=0.005, 50 iters) + 1x MSE, B=8 N=2048
//   - points resident in LDS as (x,y,z,|p|^2) float4 -> one ds_load_b128/tile
//   - rank-3 Gram cross terms computed with V_WMMA_F32_16X16X4_F32
//   - all LDS accesses via 32-bit offsets from a local extern __shared__ base
//     so address-space inference kills generic-pointer null checks
// ---------------------------------------------------------------------------

typedef float v2f __attribute__((ext_vector_type(2)));
typedef float v4f __attribute__((ext_vector_type(4)));
typedef float v8f __attribute__((ext_vector_type(8)));

#define NPTS     2048
#define NB       8
#define INV_EPS  200.0f      // 1/0.005
#define EMD_IT   50

// LDS layout (floats): [0,8191]=cloudA float4s, [8192,16383]=cloudB float4s,
//                      [16384,18431]=LU, [18432,20479]=LV   -> 80 KB total
#define A4_OFF   0           // in v4f units
#define B4_OFF   2048        // in v4f units
#define LU_OFF   16384       // in float units
#define LV_OFF   18432       // in float units
#define SMEM_FLOATS 20480

__device__ __forceinline__ v8f wmma_dot(v2f a, v2f b) {
    v8f c = {};
    // D = A(16x4) * B(4x16) + 0 ; f32 everywhere
    return __builtin_amdgcn_wmma_f32_16x16x4_f32(
        /*neg_a=*/false, a, /*neg_b=*/false, b,
        /*c_mod=*/(short)0, c, /*reuse_a=*/false, /*reuse_b=*/false);
}

__device__ __forceinline__ void wave_atomic_add(float* dst, float v) {
#pragma unroll
    for (int k = 16; k >= 1; k >>= 1) v += __shfl_xor(v, k, 32);
    if ((threadIdx.x & 31) == 0) atomicAdd(dst, v);
}

// One full sweep over the (implicit) NPTS x NPTS cost matrix C[i,j] = |p_i - q_j|^2.
// MODE 0: stores logMarg - logsumexp_j( -C/eps + LDS float at avOff+j ) to the LDS float at outOff+i (Sinkhorn half-step)
// MODE 1: returns local sum of sqrt(min_j C[i,j])                          (Chamfer direction)
// MODE 2: returns local sum of sqrt( N * sum_j exp(rl[i]-C/eps+av[j])*C )  (EMD distance)
template <int MODE>
__device__ float sweep_rows(int pOff4, int qOff4, int avOff, int outOff, int rlOff,
                            float logMarg)
{
    extern __shared__ float smemf[];
    const v4f* S4 = (const v4f*)smemf;

    const int tid  = threadIdx.x;
    const int wave = tid >> 5;
    const int lane = tid & 31;
    const int hi   = lane >> 4;     // 0: K=0,1 half   1: K=2,3 half
    const int lm   = lane & 15;
    float localSum = 0.0f;

    for (int rb = wave; rb < NPTS / 16; rb += 32) {
        const int r0 = rb << 4;
        // A tile (16 rows x K=4): K = (x, y, z, 0)
        const v4f pa = S4[pOff4 + r0 + lm];
        v2f A;
        A.x = hi ? pa.z : pa.x;
        A.y = hi ? 0.0f : pa.y;

        float anS[8], rl[8], m[8], s[8];
#pragma unroll
        for (int v = 0; v < 8; ++v) {
            const int r = r0 + v + (hi << 3);
            anS[v] = S4[pOff4 + r].w;                 // |p_r|^2 precomputed
            if (MODE == 0) { m[v] = -1e30f; s[v] = 0.0f; }
            if (MODE == 1) { m[v] = 1e30f; }
            if (MODE == 2) { s[v] = 0.0f; rl[v] = smemf[rlOff + r]; }
        }

#pragma unroll 1
        for (int jt = 0; jt < NPTS / 16; ++jt) {
            const int j0 = jt << 4;
            const v4f q = S4[qOff4 + j0 + lm];        // one ds_load_b128
            const float bn = q.w;
            v2f B;
            B.x = hi ? q.z  : q.x;
            B.y = hi ? 0.0f : q.y;
            const v8f D = wmma_dot(A, B);             // D[v] = p_{r0+v+8*hi} . q_{j0+lm}
            const float av = (MODE != 1) ? smemf[avOff + j0 + lm] : 0.0f;
#pragma unroll
            for (int v = 0; v < 8; ++v) {
                const float c = fmaxf(fmaf(D[v], -2.0f, anS[v] + bn), 0.0f);
                if (MODE == 1) {
                    m[v] = fminf(m[v], c);
                } else if (MODE == 0) {
                    const float t  = fmaf(c, -INV_EPS, av);
                    const float nm = fmaxf(m[v], t);
                    s[v] = fmaf(s[v], __expf(m[v] - nm), __expf(t - nm));
                    m[v] = nm;
                } else {
                    const float t = fmaf(c, -INV_EPS, rl[v] + av);
                    s[v] = fmaf(__expf(t), c, s[v]);
                }
            }
        }

        // reduce across the 16-lane half that shares each row
#pragma unroll
        for (int v = 0; v < 8; ++v) {
            if (MODE == 0) {
                float mv = m[v], sv = s[v];
#pragma unroll
                for (int k = 1; k <= 8; k <<= 1) {
                    const float om = __shfl_xor(mv, k, 32);
                    const float os = __shfl_xor(sv, k, 32);
                    const float nm = fmaxf(mv, om);
                    sv = fmaf(sv, __expf(mv - nm), os * __expf(om - nm));
                    mv = nm;
                }
                if (lm == 0) {
                    const int r = r0 + v + (hi << 3);
                    smemf[outOff + r] = logMarg - (mv + __logf(sv));
                }
            } else if (MODE == 1) {
                float mv = m[v];
#pragma unroll
                for (int k = 1; k <= 8; k <<= 1)
                    mv = fminf(mv, __shfl_xor(mv, k, 32));
                if (lm == 0) localSum += __fsqrt_rn(fmaxf(mv, 0.0f));
            } else {
                float sv = s[v];
#pragma unroll
                for (int k = 1; k <= 8; k <<= 1)
                    sv += __shfl_xor(sv, k, 32);
                if (lm == 0) localSum += __fsqrt_rn(fmaxf((float)NPTS * sv, 0.0f));
            }
        }
    }
    return localSum;
}

// ws layout (floats): per pair p: [4p+0]=sum sqrt(min_j C), [4p+1]=sum sqrt(min_i C),
//                     [4p+2]=sum sqrt(emd dist);  [12]=MSE sum
__global__ void corr_init_kernel(float* ws) {
    if (threadIdx.x < 16) ws[threadIdx.x] = 0.0f;
}

__global__ __launch_bounds__(1024)
void corr_pair_kernel(const float* __restrict__ shape, const float* __restrict__ p2d,
                      const float* __restrict__ srs,   const float* __restrict__ unf,
                      const float* __restrict__ srsu,  float* __restrict__ ws)
{
    extern __shared__ float smemf[];
    v4f* S4 = (v4f*)smemf;

    const int pair  = blockIdx.x >> 3;
    const int batch = blockIdx.x & 7;
    const float* aP;
    const float* bP;
    if (pair == 0)      { aP = srs;  bP = shape; }   // (self_rec_shape, shape)
    else if (pair == 1) { aP = unf;  bP = p2d;   }   // (unfold_pts, batch_p_2d)
    else                { aP = srsu; bP = shape; }   // (self_rec_shape_u, shape)
    aP += (size_t)batch * NPTS * 3;
    bP += (size_t)batch * NPTS * 3;

    for (int p = threadIdx.x; p < NPTS; p += blockDim.x) {
        const float ax = aP[3 * p], ay = aP[3 * p + 1], az = aP[3 * p + 2];
        const float bx = bP[3 * p], by = bP[3 * p + 1], bz = bP[3 * p + 2];
        v4f a4; a4.x = ax; a4.y = ay; a4.z = az; a4.w = ax * ax + ay * ay + az * az;
        v4f b4; b4.x = bx; b4.y = by; b4.z = bz; b4.w = bx * bx + by * by + bz * bz;
        S4[A4_OFF + p] = a4;
        S4[B4_OFF + p] = b4;
        smemf[LU_OFF + p] = 0.0f;
        smemf[LV_OFF + p] = 0.0f;
    }
    __syncthreads();

    // ---- Chamfer, both directions ----
    const float s1 = sweep_rows<1>(A4_OFF, B4_OFF, 0, 0, 0, 0.0f);
    const float s2 = sweep_rows<1>(B4_OFF, A4_OFF, 0, 0, 0, 0.0f);
    wave_atomic_add(&ws[pair * 4 + 0], s1);
    wave_atomic_add(&ws[pair * 4 + 1], s2);

    // ---- Sinkhorn iterations (log domain, uniform marginals) ----
    const float logMarg = -__logf((float)NPTS);
#pragma unroll 1
    for (int it = 0; it < EMD_IT; ++it) {
        sweep_rows<0>(A4_OFF, B4_OFF, LV_OFF, LU_OFF, 0, logMarg);
        __syncthreads();
        sweep_rows<0>(B4_OFF, A4_OFF, LU_OFF, LV_OFF, 0, logMarg);
        __syncthreads();
    }

    // ---- EMD: dist_i = N * sum_j exp(logu_i - C/eps + logv_j) * C ----
    const float se = sweep_rows<2>(A4_OFF, B4_OFF, LV_OFF, 0, LU_OFF, 0.0f);
    wave_atomic_add(&ws[pair * 4 + 2], se);
}

__global__ void corr_mse_kernel(const float* __restrict__ a, const float* __restrict__ b,
                                float* __restrict__ ws, int n)
{
    float acc = 0.0f;
    for (int i = blockIdx.x * blockDim.x + threadIdx.x; i < n; i += gridDim.x * blockDim.x) {
        const float d = a[i] - b[i];
        acc = fmaf(d, d, acc);
    }
    wave_atomic_add(&ws[12], acc);
}

__global__ void corr_finalize_kernel(const float* __restrict__ ws, float* __restrict__ out) {
    if (threadIdx.x == 0) {
        const float invBN = 1.0f / (float)(NB * NPTS);
        const float cd0 = (ws[0] + ws[1]) * invBN, emd0 = ws[2]  * invBN;
        const float cd1 = (ws[4] + ws[5]) * invBN, emd1 = ws[6]  * invBN;
        const float cd2 = (ws[8] + ws[9]) * invBN, emd2 = ws[10] * invBN;
        const float mse = ws[12] / (float)(NB * NPTS * 3);
        const float loss_sr     = cd0 + emd0;
        const float loss_unfold = cd1 + emd1;
        const float loss_sr_u   = mse + cd2 + emd2;
        out[0] = loss_sr + loss_unfold + loss_sr_u;
        out[1] = loss_sr;
        out[2] = loss_unfold;
        out[3] = loss_sr_u;
    }
}

extern "C" void kernel_launch(void* const* d_in, const int* in_sizes, int n_in,
                              void* d_out, int out_size, void* d_ws, size_t ws_size,
                              hipStream_t stream) {
    (void)in_sizes; (void)n_in; (void)out_size; (void)ws_size;
    const float* shape = (const float*)d_in[0];
    const float* p2d   = (const float*)d_in[1];
    const float* srs   = (const float*)d_in[2];
    const float* unf   = (const float*)d_in[3];
    const float* srsu  = (const float*)d_in[4];
    float* ws  = (float*)d_ws;
    float* out = (float*)d_out;

    corr_init_kernel<<<1, 32, 0, stream>>>(ws);
    const size_t shmem = (size_t)SMEM_FLOATS * sizeof(float);   // 80 KB of LDS
    corr_pair_kernel<<<3 * NB, 1024, shmem, stream>>>(shape, p2d, srs, unf, srsu, ws);
    corr_mse_kernel<<<64, 256, 0, stream>>>(srsu, shape, ws, NB * NPTS * 3);
    corr_finalize_kernel<<<1, 32, 0, stream>>>(ws, out);
}